// HeteroRGCNLayer_18227841204888
// MI455X (gfx1250) — compile-verified
//
#include <hip/hip_runtime.h>

typedef float v2f __attribute__((ext_vector_type(2)));
typedef float v8f __attribute__((ext_vector_type(8)));

#define D_FEAT    128
#define LDS_PITCH 132   // 128 + 4 pad: avoids 16-way LDS bank conflicts on A-frag reads

// wh[n, f] = sum_d x[n, d] * W[d, f] + bias[f]
// One wave computes one 16x16 tile via V_WMMA_F32_16X16X4_F32 (K-loop of 32 steps).
// 8 waves per block cover one 16-row stripe across all 128 output features.
__global__ void __launch_bounds__(256)
rgcn_proj_wmma(const float* __restrict__ x, const float* __restrict__ W,
               const float* __restrict__ bias, float* __restrict__ wh, int n_nodes)
{
    __shared__ float xs[16 * LDS_PITCH];

    const int row0 = blockIdx.x * 16;

    // Stage the 16x128 x-tile into LDS (coalesced, clamped for the ragged tail).
    for (int i = threadIdx.x; i < 16 * D_FEAT; i += 256) {
        const int r = i >> 7;              // / 128
        const int c = i & (D_FEAT - 1);
        int row = row0 + r;
        if (row > n_nodes - 1) row = n_nodes - 1;
        xs[r * LDS_PITCH + c] = x[(size_t)row * D_FEAT + c];
    }
    __syncthreads();

    const int wave = threadIdx.x >> 5;     // 0..7 -> 16-col tile
    const int lane = threadIdx.x & 31;
    const int col0 = wave * 16;
    const int n    = lane & 15;            // N index (== A-frag row index)
    const int kh   = (lane >> 4) << 1;     // 0 or 2: K sub-offset per half-wave

    // ISA A-layout (16x4 f32): lane t -> row (t&15), K = 2*(t>>4)+{0,1}
    // ISA B-layout (4x16 f32): lane t -> col (t&15), K = 2*(t>>4)+{0,1}
    v8f acc = {};
    for (int k0 = 0; k0 < D_FEAT; k0 += 4) {
        v2f a, b;
        a.x = xs[n * LDS_PITCH + k0 + kh];
        a.y = xs[n * LDS_PITCH + k0 + kh + 1];
        b.x = W[(size_t)(k0 + kh) * D_FEAT + col0 + n];
        b.y = W[(size_t)(k0 + kh + 1) * D_FEAT + col0 + n];
        acc = __builtin_amdgcn_wmma_f32_16x16x4_f32(
            /*neg_a=*/false, a, /*neg_b=*/false, b,
            /*c_mod=*/(short)0, acc, /*reuse_a=*/false, /*reuse_b=*/false);
    }

    // C/D layout: element (M = v + 8*(lane>>4), N = lane&15) in VGPR v.
    const float bv = bias[col0 + n];
    const int m0 = (lane >> 4) * 8;
    #pragma unroll
    for (int v = 0; v < 8; ++v) {
        const int row = row0 + m0 + v;
        if (row < n_nodes)
            wh[(size_t)row * D_FEAT + col0 + n] = acc[v] + bv;
    }
}

// out[dst[e], :] += wh[src[e], :]
// One wave per edge; each lane moves 4 consecutive floats (b128 gather,
// 4 consecutive-address f32 atomics -> L2-resident RMW).
__global__ void __launch_bounds__(256)
rgcn_scatter(const float* __restrict__ wh, const int* __restrict__ src,
             const int* __restrict__ dst, float* __restrict__ out, int n_edges)
{
    const int e = (int)((blockIdx.x * 256u + threadIdx.x) >> 5);
    if (e >= n_edges) return;
    const int lane = threadIdx.x & 31;
    const int s = src[e];
    const int d = dst[e];
    const float4 v = *(const float4*)(wh + (size_t)s * D_FEAT + lane * 4);
    float* o = out + (size_t)d * D_FEAT + lane * 4;
    atomicAdd(o + 0, v.x);
    atomicAdd(o + 1, v.y);
    atomicAdd(o + 2, v.z);
    atomicAdd(o + 3, v.w);
}

extern "C" void kernel_launch(void* const* d_in, const int* in_sizes, int n_in,
                              void* d_out, int out_size, void* d_ws, size_t ws_size,
                              hipStream_t stream)
{
    const float* x   = (const float*)d_in[0];
    const float* W   = (const float*)d_in[1];
    const float* b   = (const float*)d_in[2];
    const int*   src = (const int*)d_in[3];
    const int*   dst = (const int*)d_in[4];
    float* out = (float*)d_out;
    float* wh  = (float*)d_ws;    // one relation's projection: n_nodes*128 floats

    const int n_nodes = in_sizes[0] / D_FEAT;          // 50000
    const int n_rel   = in_sizes[2] / D_FEAT;          // 3
    const int n_edges = in_sizes[3] / n_rel;           // 600000

    hipMemsetAsync(d_out, 0, (size_t)out_size * sizeof(float), stream);

    const int row_tiles   = (n_nodes + 15) / 16;
    const int scat_blocks = (n_edges * 32 + 255) / 256;

    for (int r = 0; r < n_rel; ++r) {
        rgcn_proj_wmma<<<row_tiles, 256, 0, stream>>>(
            x, W + (size_t)r * D_FEAT * D_FEAT, b + (size_t)r * D_FEAT, wh, n_nodes);
        rgcn_scatter<<<scat_blocks, 256, 0, stream>>>(
            wh, src + (size_t)r * n_edges, dst + (size_t)r * n_edges, out, n_edges);
    }
}